// LightGCNBert_61632780698014
// MI455X (gfx1250) — compile-verified
//
#include <hip/hip_runtime.h>

#define K_USERS 200000
#define K_ITEMS 100000
#define K_NODES 300000
#define K_EDGES 1500000
#define K_DIM   64
#define K_BERT  384
#define K_LAYERS 3

typedef __attribute__((ext_vector_type(2))) float v2f;
typedef __attribute__((ext_vector_type(8))) float v8f;

// ---------------------------------------------------------------------------
// users: u0 = user_emb + gender_emb[g] + age_emb[a]  -> x[0:U*64) and acc
// ---------------------------------------------------------------------------
__global__ void lgcn_user_init(const float* __restrict__ user_emb,
                               const float* __restrict__ gender_emb,
                               const float* __restrict__ age_emb,
                               const int* __restrict__ user_gender,
                               const int* __restrict__ user_age,
                               float* __restrict__ x,
                               float* __restrict__ acc) {
  int idx = blockIdx.x * blockDim.x + threadIdx.x;
  if (idx >= K_USERS * K_DIM) return;
  int n = idx >> 6, d = idx & 63;
  float v = user_emb[idx]
          + gender_emb[user_gender[n] * K_DIM + d]
          + age_emb[user_age[n] * K_DIM + d];
  x[idx]   = v;
  acc[idx] = v;
}

// ---------------------------------------------------------------------------
// items: i0 = item_emb + cat_emb[cat] + item_bert @ W^T
// WMMA f32 16x16x4. One wave per 16 item rows, computes all 64 output cols
// (4 column tiles) so item_bert (154 MB) is streamed exactly once.
//
// A (16x4 f32):  lane[0:15] holds M=lane, K={k,k+1}; lane[16:31] K={k+2,k+3}
// B (4x16 f32):  lane[0:15] holds N=lane, K={k,k+1}; lane[16:31] K={k+2,k+3}
//                B[kk,n] = W[d0+n, k+kk]  (W row-major [64,384], K contiguous)
// C/D (16x16 f32, 8 VGPRs): vgpr j -> M = j + 8*(lane>=16), N = lane&15
// ---------------------------------------------------------------------------
__global__ void lgcn_item_init_wmma(const float* __restrict__ bert,   // [I, 384]
                                    const float* __restrict__ w,      // [64, 384]
                                    const float* __restrict__ item_emb,
                                    const float* __restrict__ cat_emb,
                                    const int* __restrict__ item_cat,
                                    float* __restrict__ x,
                                    float* __restrict__ acc) {
  const int wave = (blockIdx.x * blockDim.x + threadIdx.x) >> 5;
  const int lane = threadIdx.x & 31;
  const int i0 = wave * 16;
  if (i0 >= K_ITEMS) return;
  const int row  = lane & 15;
  const int koff = (lane >> 4) * 2;   // K-pair offset for this half-wave
  const int hi8  = (lane >> 4) * 8;   // C/D row offset for this half-wave

  const float* ap  = bert + (size_t)(i0 + row) * K_BERT + koff;
  const float* bp0 = w + (size_t)(row)      * K_BERT + koff;
  const float* bp1 = w + (size_t)(16 + row) * K_BERT + koff;
  const float* bp2 = w + (size_t)(32 + row) * K_BERT + koff;
  const float* bp3 = w + (size_t)(48 + row) * K_BERT + koff;

  v8f c0 = {}, c1 = {}, c2 = {}, c3 = {};
#pragma unroll 8
  for (int k = 0; k < K_BERT; k += 4) {
    v2f a  = *(const v2f*)(ap  + k);
    v2f b0 = *(const v2f*)(bp0 + k);
    v2f b1 = *(const v2f*)(bp1 + k);
    v2f b2 = *(const v2f*)(bp2 + k);
    v2f b3 = *(const v2f*)(bp3 + k);
    c0 = __builtin_amdgcn_wmma_f32_16x16x4_f32(false, a, false, b0, (short)0, c0, false, false);
    c1 = __builtin_amdgcn_wmma_f32_16x16x4_f32(false, a, false, b1, (short)0, c1, false, false);
    c2 = __builtin_amdgcn_wmma_f32_16x16x4_f32(false, a, false, b2, (short)0, c2, false, false);
    c3 = __builtin_amdgcn_wmma_f32_16x16x4_f32(false, a, false, b3, (short)0, c3, false, false);
  }

  // epilogue: add item_emb + cat_emb[item_cat], write x and acc (node space)
#pragma unroll
  for (int j = 0; j < 8; ++j) {
    const int gRow = i0 + hi8 + j;
    const int cat  = item_cat[gRow];
    const size_t rbase = (size_t)gRow * K_DIM;
    const size_t obase = (size_t)(K_USERS + gRow) * K_DIM;
    int col; float v;
    col = row;      v = c0[j] + item_emb[rbase + col] + cat_emb[cat * K_DIM + col];
    x[obase + col] = v; acc[obase + col] = v;
    col = 16 + row; v = c1[j] + item_emb[rbase + col] + cat_emb[cat * K_DIM + col];
    x[obase + col] = v; acc[obase + col] = v;
    col = 32 + row; v = c2[j] + item_emb[rbase + col] + cat_emb[cat * K_DIM + col];
    x[obase + col] = v; acc[obase + col] = v;
    col = 48 + row; v = c3[j] + item_emb[rbase + col] + cat_emb[cat * K_DIM + col];
    x[obase + col] = v; acc[obase + col] = v;
  }
}

// ---------------------------------------------------------------------------
__global__ void lgcn_zero(float4* __restrict__ p, int n4) {
  int i = blockIdx.x * blockDim.x + threadIdx.x;
  if (i < n4) p[i] = make_float4(0.f, 0.f, 0.f, 0.f);
}

// COO SpMM: one wave32 per edge, 2 dims per lane (float2 gather + 2 f32 atomics)
__global__ void lgcn_spmm(const float* __restrict__ adj_val,
                          const int* __restrict__ adj_row,
                          const int* __restrict__ adj_col,
                          const float* __restrict__ xin,
                          float* __restrict__ yout) {
  const int e = (blockIdx.x * blockDim.x + threadIdx.x) >> 5;
  if (e >= K_EDGES) return;
  const int lane = threadIdx.x & 31;
  const float val = adj_val[e];
  const int r = adj_row[e];
  const int c = adj_col[e];
  const v2f xv = *(const v2f*)(xin + (size_t)c * K_DIM + lane * 2);
  float* yp = yout + (size_t)r * K_DIM + lane * 2;
  unsafeAtomicAdd(yp,     val * xv.x);
  unsafeAtomicAdd(yp + 1, val * xv.y);
}

// acc = (acc + y) * scale   (scale folds the final /4 into the last layer)
__global__ void lgcn_acc(const float4* __restrict__ y, float4* __restrict__ out,
                         int n4, float scale) {
  int i = blockIdx.x * blockDim.x + threadIdx.x;
  if (i >= n4) return;
  float4 o = out[i];
  float4 v = y[i];
  o.x = (o.x + v.x) * scale;
  o.y = (o.y + v.y) * scale;
  o.z = (o.z + v.z) * scale;
  o.w = (o.w + v.w) * scale;
  out[i] = o;
}

// ---------------------------------------------------------------------------
extern "C" void kernel_launch(void* const* d_in, const int* in_sizes, int n_in,
                              void* d_out, int out_size, void* d_ws, size_t ws_size,
                              hipStream_t stream) {
  const float* user_emb    = (const float*)d_in[0];
  const float* item_emb    = (const float*)d_in[1];
  const float* gender_emb  = (const float*)d_in[2];
  const float* age_emb     = (const float*)d_in[3];
  const float* cat_emb     = (const float*)d_in[4];
  const float* bert_proj_w = (const float*)d_in[5];   // [64, 384]
  const float* item_bert   = (const float*)d_in[6];   // [100000, 384]
  const float* adj_val     = (const float*)d_in[7];
  const int*   user_gender = (const int*)d_in[8];
  const int*   user_age    = (const int*)d_in[9];
  const int*   item_cat    = (const int*)d_in[10];
  const int*   adj_row     = (const int*)d_in[11];
  const int*   adj_col     = (const int*)d_in[12];
  float* acc = (float*)d_out;                          // [300000, 64]

  float* x = (float*)d_ws;                             // ping
  float* y = x + (size_t)K_NODES * K_DIM;              // pong

  // --- layer-0 embeddings into x and acc ---
  lgcn_user_init<<<(K_USERS * K_DIM + 255) / 256, 256, 0, stream>>>(
      user_emb, gender_emb, age_emb, user_gender, user_age, x, acc);

  const int gemm_threads = (K_ITEMS / 16) * 32;        // one wave / 16 rows
  lgcn_item_init_wmma<<<(gemm_threads + 255) / 256, 256, 0, stream>>>(
      item_bert, bert_proj_w, item_emb, cat_emb, item_cat, x, acc);

  // --- 3 propagation layers ---
  const int n4 = K_NODES * K_DIM / 4;
  float* cur = x;
  float* nxt = y;
  for (int L = 0; L < K_LAYERS; ++L) {
    lgcn_zero<<<(n4 + 255) / 256, 256, 0, stream>>>((float4*)nxt, n4);
    lgcn_spmm<<<(K_EDGES * 32 + 255) / 256, 256, 0, stream>>>(
        adj_val, adj_row, adj_col, cur, nxt);
    lgcn_acc<<<(n4 + 255) / 256, 256, 0, stream>>>(
        (const float4*)nxt, (float4*)acc, n4,
        (L == K_LAYERS - 1) ? 0.25f : 1.0f);
    float* t = cur; cur = nxt; nxt = t;
  }
}